// Sparsemax_60138132078893
// MI455X (gfx1250) — compile-verified
//
#include <hip/hip_runtime.h>
#include <stdint.h>

// Sparsemax over rows of K=2048 fp32, one 256-thread block per row.
// Memory-bound: 256 MiB total traffic -> ~11us floor at 23.3 TB/s.
// CDNA5 paths: TDM tensor_load_to_lds (8KB row DMA into LDS), s_wait_tensorcnt,
// ds_load_b128 from 16B-aligned LDS, wave32 shuffle reductions, NT b128 stores.

typedef __attribute__((ext_vector_type(4))) unsigned int v4u_t;
typedef __attribute__((ext_vector_type(8))) int         v8i_t;
typedef __attribute__((ext_vector_type(4))) int         v4i_t;
typedef __attribute__((ext_vector_type(4))) float       v4f_t;

#define SPX_K       2048
#define SPX_THREADS 256
#define SPX_WAVES   (SPX_THREADS / 32)

// Reduce (s, c) over the whole block. Wave32 shuffle tree, then LDS across waves.
__device__ __forceinline__ void spx_block_reduce2(float& s, float& c,
                                                  float* rs, float* rc) {
#pragma unroll
  for (int off = 16; off > 0; off >>= 1) {
    s += __shfl_xor(s, off, 32);
    c += __shfl_xor(c, off, 32);
  }
  const int wave = threadIdx.x >> 5;
  const int lane = threadIdx.x & 31;
  if (lane == 0) { rs[wave] = s; rc[wave] = c; }
  __syncthreads();
  float ts = 0.0f, tc = 0.0f;
#pragma unroll
  for (int w = 0; w < SPX_WAVES; ++w) { ts += rs[w]; tc += rc[w]; }
  __syncthreads();   // protect scratch before next iteration's writes
  s = ts;
  c = tc;
}

__global__ __launch_bounds__(SPX_THREADS)
void sparsemax2048_kernel(const float* __restrict__ in, float* __restrict__ out) {
  // Only dynamic LDS is used => its base byte offset within the WG allocation
  // is 0, which is what the TDM descriptor's lds_addr field needs.
  extern __shared__ __attribute__((aligned(16))) float smem[];  // [0..K-1] row
  float* rs = smem + SPX_K;                // [K .. K+7]    per-wave partial sums
  float* rc = smem + SPX_K + SPX_WAVES;    // [K+8 .. K+15] per-wave partial counts

  const int row = blockIdx.x;
  const float* src = in + (size_t)row * SPX_K;

  // ---- TDM: DMA the full 8KB row Global -> LDS (issued once, by wave 0) ----
  if (threadIdx.x == 0) {
    const uint64_t ga = (uint64_t)(uintptr_t)src;
    // D# group 0 (128b): count=1 valid descriptor, lds_addr=0,
    // global_addr[56:0] in bits [120:64], type=2 ("image") in bits [127:126].
    v4u_t g0;
    g0[0] = 1u;
    g0[1] = 0u;
    g0[2] = (unsigned)(ga & 0xFFFFFFFFu);
    g0[3] = (unsigned)((ga >> 32) & 0x01FFFFFFu) | (2u << 30);
    // D# group 1 (256b): data_size=4B, tensor 2048x1, tile 2048x1,
    // tensor_dim0_stride=2048; no multicast / barrier / pad / iterate.
    v8i_t g1;
    g1[0] = (int)(2u << 16);        // data_size = 2 -> 4 bytes
    g1[1] = (int)(2048u << 16);     // tensor_dim0[15:0] = 2048 (hi bits 0)
    g1[2] = (int)(1u << 16);        // tensor_dim0[31:16]=0, tensor_dim1 = 1
    g1[3] = (int)(2048u << 16);     // tensor_dim1 hi=0, tile_dim0 = 2048
    g1[4] = 1;                      // tile_dim1 = 1, tile_dim2 = 0 (2D tile)
    g1[5] = 2048;                   // tensor_dim0_stride[31:0]
    g1[6] = 0;                      // stride0 hi, stride1 lo
    g1[7] = 0;
    v4i_t gz = {0, 0, 0, 0};        // groups 2/3 unused (<=2D tensor)
#if __clang_major__ >= 23
    v8i_t gz8 = {0, 0, 0, 0, 0, 0, 0, 0};
    __builtin_amdgcn_tensor_load_to_lds(g0, g1, gz, gz, gz8, 0);
#else
    __builtin_amdgcn_tensor_load_to_lds(g0, g1, gz, gz, 0);
#endif
    __builtin_amdgcn_s_wait_tensorcnt(0);
  }
  __syncthreads();

  // ---- Pull this thread's 8 elements into registers (2x ds_load_b128) ----
  const int tid = threadIdx.x;
  const v4f_t a = ((const v4f_t*)smem)[tid];
  const v4f_t b = ((const v4f_t*)smem)[tid + SPX_THREADS];

  // ---- tau0 = (sum(z) - 1) / K ----
  float s = ((a[0] + a[1]) + (a[2] + a[3])) + ((b[0] + b[1]) + (b[2] + b[3]));
  float c = 0.0f;
  spx_block_reduce2(s, c, rs, rc);

  float tau      = (s - 1.0f) * (1.0f / (float)SPX_K);
  float prev_cnt = (float)SPX_K;

  // ---- Michelot fixed point: tau <- (sum_{z>tau} z - 1) / |{z>tau}| ----
  // Support shrinks monotonically; converges exactly, typically < 10 iters.
  for (int it = 0; it < 48; ++it) {
    float ps = 0.0f, pc = 0.0f;
#pragma unroll
    for (int i = 0; i < 4; ++i) {
      if (a[i] > tau) { ps += a[i]; pc += 1.0f; }
      if (b[i] > tau) { ps += b[i]; pc += 1.0f; }
    }
    spx_block_reduce2(ps, pc, rs, rc);
    const float ntau = (ps - 1.0f) / pc;   // pc >= 1 always (max elem > tau)
    const bool  done = (pc == prev_cnt);   // support stable -> tau is exact
    tau      = ntau;
    prev_cnt = pc;
    if (done) break;                       // uniform across the block
  }

  // ---- out = max(z - tau, 0), non-temporal 128-bit stores ----
  v4f_t oa, ob;
#pragma unroll
  for (int i = 0; i < 4; ++i) {
    oa[i] = fmaxf(a[i] - tau, 0.0f);
    ob[i] = fmaxf(b[i] - tau, 0.0f);
  }
  v4f_t* dst = (v4f_t*)(out + (size_t)row * SPX_K);
  __builtin_nontemporal_store(oa, dst + tid);
  __builtin_nontemporal_store(ob, dst + tid + SPX_THREADS);
}

extern "C" void kernel_launch(void* const* d_in, const int* in_sizes, int n_in,
                              void* d_out, int out_size, void* d_ws, size_t ws_size,
                              hipStream_t stream) {
  (void)n_in; (void)out_size; (void)d_ws; (void)ws_size;
  const float* in  = (const float*)d_in[0];
  float*       out = (float*)d_out;
  const int rows = in_sizes[0] / SPX_K;           // (4,4096,2048) -> 16384 rows
  const size_t smem_bytes = (SPX_K + 2 * SPX_WAVES) * sizeof(float);
  sparsemax2048_kernel<<<rows, SPX_THREADS, smem_bytes, stream>>>(in, out);
}